// LightGCN_86646670229545
// MI455X (gfx1250) — compile-verified
//
#include <hip/hip_runtime.h>
#include <stdint.h>

#define NU      100000
#define NI      50000
#define NNODES  150000
#define DIM     64
#define LAYERS  3
#define EPSV    1e-8f

#ifndef __has_builtin
#define __has_builtin(x) 0
#endif

#if __has_builtin(__builtin_amdgcn_global_load_async_to_lds_b128)
#define USE_ASYNC_LDS 1
#else
#define USE_ASYNC_LDS 0
#endif

// exact parameter types for the async-LDS builtin (from clang diagnostic):
// arg0: global (AS1) pointer to 4-int vector, arg1: LDS (AS3) pointer to 4-int vector
typedef int v4i __attribute__((ext_vector_type(4)));
typedef __attribute__((address_space(1))) v4i* gv4i_p;
typedef __attribute__((address_space(3))) v4i* lv4i_p;

__device__ __forceinline__ void atom_add_f32(float* p, float v) {
    // agent-scope relaxed fp32 add -> global_atomic_add_f32 (non-returning)
    (void)__hip_atomic_fetch_add(p, v, __ATOMIC_RELAXED, __HIP_MEMORY_SCOPE_AGENT);
}

__device__ __forceinline__ void wait_async_all() {
#if USE_ASYNC_LDS
#if __has_builtin(__builtin_amdgcn_s_wait_asynccnt)
    __builtin_amdgcn_s_wait_asynccnt(0);
#else
    asm volatile("s_wait_asynccnt 0" ::: "memory");
#endif
#endif
}

// ---------------- elementwise kernels ----------------

__global__ void k_zero4(float4* __restrict__ p, int n4) {
    int i = blockIdx.x * blockDim.x + threadIdx.x;
    if (i < n4) p[i] = make_float4(0.f, 0.f, 0.f, 0.f);
}

// x = acc = concat(user_emb, item_emb); also zero the first scatter target
__global__ void k_init(const float4* __restrict__ ue, const float4* __restrict__ ie,
                       float4* __restrict__ x, float4* __restrict__ acc,
                       float4* __restrict__ z) {
    const int userN4 = NU * DIM / 4;      // 1,600,000
    const int totN4  = NNODES * DIM / 4;  // 2,400,000
    int i = blockIdx.x * blockDim.x + threadIdx.x;
    if (i < totN4) {
        float4 v = (i < userN4) ? ue[i] : ie[i - userN4];
        x[i] = v;
        acc[i] = v;
        z[i] = make_float4(0.f, 0.f, 0.f, 0.f);
    }
}

__global__ void k_deg(const int* __restrict__ rows, const int* __restrict__ cols,
                      float* __restrict__ deg, int nnz) {
    int e = blockIdx.x * blockDim.x + threadIdx.x;
    if (e < nnz) {
        __builtin_prefetch(rows + e + 8192, 0, 0);   // global_prefetch_b8 (speculative)
        __builtin_prefetch(cols + e + 8192, 0, 0);
        atom_add_f32(deg + rows[e], 1.0f);
        atom_add_f32(deg + NU + cols[e], 1.0f);
    }
}

__global__ void k_invsq(float* __restrict__ deg, int n) {
    int i = blockIdx.x * blockDim.x + threadIdx.x;
    if (i < n) deg[i] = rsqrtf(deg[i] + EPSV);
}

// acc = (acc + x) * scale; optionally zero the retiring buffer (next scatter target)
__global__ void k_accum_zero(float4* __restrict__ acc, const float4* __restrict__ x,
                             float4* __restrict__ z, int n4, float scale, int doZero) {
    int i = blockIdx.x * blockDim.x + threadIdx.x;
    if (i < n4) {
        float4 a = acc[i];
        float4 b = x[i];
        a.x = (a.x + b.x) * scale;
        a.y = (a.y + b.y) * scale;
        a.z = (a.z + b.z) * scale;
        a.w = (a.w + b.w) * scale;
        acc[i] = a;
        if (doZero) z[i] = make_float4(0.f, 0.f, 0.f, 0.f);
    }
}

// ---------------- SpMM: one wave per edge, both directions ----------------

#define TILE         512          // edges per LDS tile
#define SPMM_THREADS 256          // 8 waves per block
#define SPMM_BLOCKS  1024
#define NW           (SPMM_THREADS / 32)

__global__ __launch_bounds__(SPMM_THREADS)
void k_spmm(const int* __restrict__ rows, const int* __restrict__ cols,
            const float* __restrict__ invsq,
            const float* __restrict__ xin, float* __restrict__ xout, int nnz)
{
    __shared__ int s_rows[2][TILE];
    __shared__ int s_cols[2][TILE];

    const int tid   = threadIdx.x;
    const int wave  = tid >> 5;
    const int lane  = tid & 31;
    const int dir   = lane >> 4;          // 0: item->user accumulate, 1: user->item
    const int chunk = (lane & 15) << 2;   // float offset inside the 64-float row

    // stage one 512-edge tile (128 threads x 16B for rows, same for cols)
    auto stage = [&](int base, int b) {
        if (tid < TILE / 4) {
            int i0 = base + tid * 4;
            if (i0 + 3 < nnz) {
#if USE_ASYNC_LDS
                __builtin_amdgcn_global_load_async_to_lds_b128(
                    (gv4i_p)(rows + i0), (lv4i_p)&s_rows[b][tid * 4], 0, 0);
                __builtin_amdgcn_global_load_async_to_lds_b128(
                    (gv4i_p)(cols + i0), (lv4i_p)&s_cols[b][tid * 4], 0, 0);
#else
                *(int4*)&s_rows[b][tid * 4] = *(const int4*)(rows + i0);
                *(int4*)&s_cols[b][tid * 4] = *(const int4*)(cols + i0);
#endif
            } else {
                for (int k = 0; k < 4; ++k) {
                    int i = i0 + k;
                    if (i < nnz) {
                        s_rows[b][tid * 4 + k] = rows[i];
                        s_cols[b][tid * 4 + k] = cols[i];
                    }
                }
            }
        }
    };

    const int tile0 = blockIdx.x * TILE;
    const int tstep = gridDim.x * TILE;

    if (tile0 < nnz) stage(tile0, 0);
    int buf = 0;

    for (int t = tile0; t < nnz; t += tstep) {
        wait_async_all();          // current tile's async stage complete (issuing waves)
        __syncthreads();           // publish LDS to all waves

        int tn = t + tstep;        // kick off next tile into the other buffer
        if (tn < nnz) stage(tn, buf ^ 1);

        int cnt = nnz - t; if (cnt > TILE) cnt = TILE;

        // 2x unrolled: issue both gathers before the atomics for more MLP
        for (int e = wave; e < cnt; e += 2 * NW) {
            int e2 = e + NW;
            bool has2 = (e2 < cnt);

            int u1 = s_rows[buf][e];
            int c1 = s_cols[buf][e] + NU;
            int u2 = has2 ? s_rows[buf][e2] : u1;
            int c2 = has2 ? s_cols[buf][e2] + NU : c1;

            float w1 = invsq[u1] * invsq[c1];
            float w2 = invsq[u2] * invsq[c2];

            int s1 = dir ? c1 : u1, d1 = dir ? u1 : c1;
            int s2 = dir ? c2 : u2, d2 = dir ? u2 : c2;

            const float4 v1 = *(const float4*)(xin + d1 * DIM + chunk);
            const float4 v2 = *(const float4*)(xin + d2 * DIM + chunk);

            float* o1 = xout + s1 * DIM + chunk;
            atom_add_f32(o1 + 0, w1 * v1.x);
            atom_add_f32(o1 + 1, w1 * v1.y);
            atom_add_f32(o1 + 2, w1 * v1.z);
            atom_add_f32(o1 + 3, w1 * v1.w);

            if (has2) {
                float* o2 = xout + s2 * DIM + chunk;
                atom_add_f32(o2 + 0, w2 * v2.x);
                atom_add_f32(o2 + 1, w2 * v2.y);
                atom_add_f32(o2 + 2, w2 * v2.z);
                atom_add_f32(o2 + 3, w2 * v2.w);
            }
        }
        __syncthreads();           // everyone done reading buf before it is restaged
        buf ^= 1;
    }
}

// ---------------- host launcher ----------------

extern "C" void kernel_launch(void* const* d_in, const int* in_sizes, int n_in,
                              void* d_out, int out_size, void* d_ws, size_t ws_size,
                              hipStream_t stream) {
    const float* ue  = (const float*)d_in[0];
    const float* ie  = (const float*)d_in[1];
    const int* rows  = (const int*)d_in[2];
    const int* cols  = (const int*)d_in[3];
    float* out       = (float*)d_out;
    const int nnz    = in_sizes[2];

    // workspace layout: deg/invsq (150016 f32, padded), then two ping-pong feature buffers
    char* ws  = (char*)d_ws;
    float* deg = (float*)ws;
    const size_t degBytes = (size_t)150016 * sizeof(float);       // 600,064 B
    const size_t xBytes   = (size_t)NNODES * DIM * sizeof(float); // 38,400,000 B
    float* xA = (float*)(ws + degBytes);
    float* xB = (float*)(ws + degBytes + xBytes);

    const int totN4 = NNODES * DIM / 4;   // 2,400,000
    const int degN4 = 150016 / 4;         // 37,504

    k_zero4<<<(degN4 + 255) / 256, 256, 0, stream>>>((float4*)deg, degN4);
    // x_cur = acc = embeddings; zero first scatter target (xB)
    k_init <<<(totN4 + 255) / 256, 256, 0, stream>>>(
        (const float4*)ue, (const float4*)ie, (float4*)xA, (float4*)out, (float4*)xB);
    k_deg  <<<(nnz + 255) / 256, 256, 0, stream>>>(rows, cols, deg, nnz);
    k_invsq<<<(NNODES + 255) / 256, 256, 0, stream>>>(deg, NNODES);

    float* cur = xA;
    float* nxt = xB;
    for (int l = 0; l < LAYERS; ++l) {
        k_spmm<<<SPMM_BLOCKS, SPMM_THREADS, 0, stream>>>(rows, cols, deg, cur, nxt, nnz);
        // acc = (acc + nxt) * scale; zero `cur` (it becomes next layer's scatter target)
        k_accum_zero<<<(totN4 + 255) / 256, 256, 0, stream>>>(
            (float4*)out, (const float4*)nxt, (float4*)cur, totN4,
            (l == LAYERS - 1) ? 0.25f : 1.0f, (l == LAYERS - 1) ? 0 : 1);
        float* tmp = cur; cur = nxt; nxt = tmp;
    }
}